// GNNClassifier_661424964180
// MI455X (gfx1250) — compile-verified
//
#include <hip/hip_runtime.h>

typedef __attribute__((ext_vector_type(2))) float v2f;
typedef __attribute__((ext_vector_type(8))) float v8f;

// ---------------------------------------------------------------- utilities

__global__ void fill_val_kernel(float* __restrict__ p, float v, int n) {
    int i = blockIdx.x * blockDim.x + threadIdx.x;
    if (i < n) p[i] = v;
}

// deg[dst[e]] += 1.0f  (deg pre-initialized to 1.0 for the self loop)
__global__ void deg_count_kernel(const long long* __restrict__ dst,
                                 float* __restrict__ deg, int E) {
    int e = blockIdx.x * blockDim.x + threadIdx.x;
    if (e < E) atomicAdd(&deg[(int)dst[e]], 1.0f);
}

// p = (p > 0) ? rsqrt(p) : 0   (in place: deg -> dinv)
__global__ void rsqrt_ip_kernel(float* __restrict__ p, int n) {
    int i = blockIdx.x * blockDim.x + threadIdx.x;
    if (i < n) {
        float d = p[i];
        p[i] = (d > 0.0f) ? rsqrtf(d) : 0.0f;
    }
}

// norm[e] = dinv[src[e]] * dinv[dst[e]]
__global__ void edge_norm_kernel(const long long* __restrict__ src,
                                 const long long* __restrict__ dst,
                                 const float* __restrict__ dinv,
                                 float* __restrict__ norm, int E) {
    int e = blockIdx.x * blockDim.x + threadIdx.x;
    if (e < E) norm[e] = dinv[(int)src[e]] * dinv[(int)dst[e]];
}

// out[i][c] = h[i][c] * dinv[i]^2 (+ bias[c]), float4-vectorized
__global__ void self_loop_init_kernel(const float4* __restrict__ h4,
                                      const float* __restrict__ dinv,
                                      const float4* __restrict__ bias4,
                                      float4* __restrict__ out4,
                                      int n_nodes, int C4) {
    int idx = blockIdx.x * blockDim.x + threadIdx.x;
    if (idx >= n_nodes * C4) return;
    int i = idx / C4;
    int c4 = idx - i * C4;
    float w = dinv[i];
    w *= w;
    float4 v = h4[idx];
    v.x *= w; v.y *= w; v.z *= w; v.w *= w;
    if (bias4) {
        float4 b = bias4[c4];
        v.x += b.x; v.y += b.y; v.z += b.z; v.w += b.w;
    }
    out4[idx] = v;
}

// One WAVE per edge: scalar index loads (readfirstlane -> SMEM), lanes stream
// coalesced float4 of the source row, non-returning f32 atomics to dst row.
__global__ void scatter_edges_kernel(const long long* __restrict__ src,
                                     const long long* __restrict__ dst,
                                     const float* __restrict__ norm,
                                     const float* __restrict__ h,
                                     float* __restrict__ out,
                                     int E, int C) {
    int wave = threadIdx.x >> 5;
    int lane = threadIdx.x & 31;
    int e = __builtin_amdgcn_readfirstlane(blockIdx.x * (blockDim.x >> 5) + wave);
    if (e >= E) return;
    int   s = (int)src[e];          // wave-uniform
    int   d = (int)dst[e];          // wave-uniform
    float w = norm[e];              // wave-uniform
    const float* hs = h   + (size_t)s * C;
    float*       od = out + (size_t)d * C;
    for (int c = lane * 4; c < C; c += 128) {
        const float4 v = *(const float4*)(hs + c);
        atomicAdd(od + c + 0, v.x * w);
        atomicAdd(od + c + 1, v.y * w);
        atomicAdd(od + c + 2, v.z * w);
        atomicAdd(od + c + 3, v.w * w);
    }
}

// ------------------------------------------------------------- WMMA GEMM f32
// Cm[M,N] = op(A)[M,K] @ Bm[K,N], op = identity or relu(x + kbias[k]).
// One wave computes a 16x64 tile (4 accumulators) with V_WMMA_F32_16X16X4_F32;
// the A fragment is loaded once per k-step and feeds 4 WMMAs.
// A frag (16x4 f32): lanes 0-15 hold {K=0,1}, lanes 16-31 hold {K=2,3}, M=lane&15.
// B frag (4x16 f32): VGPR0 = row K=(half*2), VGPR1 = row K=(half*2+1), N=lane&15.
// C/D (16x16 f32, 8 VGPRs): VGPR r -> M = r + 8*half, N = lane&15.
template <bool RELU_BIAS>
__global__ __launch_bounds__(128) void gemm_wmma_f32_kernel(
        const float* __restrict__ A,
        const float* __restrict__ Bm,
        const float* __restrict__ kbias,   // per-K bias (used when RELU_BIAS)
        float* __restrict__ Cm,
        int M, int K, int N) {
    const int wave = threadIdx.x >> 5;     // 4 waves per block
    const int lane = threadIdx.x & 31;
    const int half = lane >> 4;            // 0 or 1
    const int l15  = lane & 15;

    const int nt64   = N >> 6;             // 64-col tiles across N
    const int mtiles = (M + 15) >> 4;
    const int wid    = blockIdx.x * 4 + wave;
    int mtile = wid / nt64;
    const int n64 = wid - mtile * nt64;    // always in [0, nt64)
    if (mtile >= mtiles) mtile = mtiles - 1;   // spare waves: duplicate work (benign)

    int row = mtile * 16 + l15;
    if (row >= M) row = M - 1;             // clamp: keep EXEC all-1s for WMMA
    const int col0 = n64 * 64 + l15;
    const int kq   = half * 2;

    const float* ap  = A + (size_t)row * K + kq;
    const float* bp0 = Bm + (size_t)kq * N + col0;
    const float* bp1 = bp0 + N;
    const size_t bstep = (size_t)4 * N;

    v8f acc0 = {}, acc1 = {}, acc2 = {}, acc3 = {};

#pragma unroll 2
    for (int k = 0; k < K; k += 4) {
        float a0 = ap[0];
        float a1 = ap[1];
        if (RELU_BIAS) {
            a0 = fmaxf(a0 + kbias[k + kq],     0.0f);
            a1 = fmaxf(a1 + kbias[k + kq + 1], 0.0f);
        }
        const v2f av  = {a0, a1};
        const v2f bv0 = {bp0[0],  bp1[0]};
        const v2f bv1 = {bp0[16], bp1[16]};
        const v2f bv2 = {bp0[32], bp1[32]};
        const v2f bv3 = {bp0[48], bp1[48]};
        acc0 = __builtin_amdgcn_wmma_f32_16x16x4_f32(false, av, false, bv0, (short)0, acc0, false, false);
        acc1 = __builtin_amdgcn_wmma_f32_16x16x4_f32(false, av, false, bv1, (short)0, acc1, false, false);
        acc2 = __builtin_amdgcn_wmma_f32_16x16x4_f32(false, av, false, bv2, (short)0, acc2, false, false);
        acc3 = __builtin_amdgcn_wmma_f32_16x16x4_f32(false, av, false, bv3, (short)0, acc3, false, false);
        ap  += 4;
        bp0 += bstep;
        bp1 += bstep;
    }

    const int mbase = mtile * 16 + half * 8;
#pragma unroll
    for (int r = 0; r < 8; ++r) {
        const int m = mbase + r;
        if (m < M) {
            float* crow = Cm + (size_t)m * N + col0;
            crow[0]  = acc0[r];
            crow[16] = acc1[r];
            crow[32] = acc2[r];
            crow[48] = acc3[r];
        }
    }
}

// ---------------------------------------------------------------- launcher

static inline size_t align_up(size_t v, size_t a) { return (v + a - 1) & ~(a - 1); }

extern "C" void kernel_launch(void* const* d_in, const int* in_sizes, int n_in,
                              void* d_out, int out_size, void* d_ws, size_t ws_size,
                              hipStream_t stream) {
    const float*     x   = (const float*)d_in[0];
    const long long* ei  = (const long long*)d_in[1];   // int64 [2, E]
    const float*     W1  = (const float*)d_in[2];
    const float*     b1  = (const float*)d_in[3];
    const float*     W2  = (const float*)d_in[4];
    const float*     b2  = (const float*)d_in[5];
    float*           out = (float*)d_out;

    const int HID   = in_sizes[3];                // 256
    const int IN_C  = in_sizes[2] / HID;          // 256
    const int OUT_C = in_sizes[5];                // 128
    const int NN    = in_sizes[0] / IN_C;         // 50000 nodes
    const int E     = in_sizes[1] / 2;            // 800000 edges

    const long long* src = ei;
    const long long* dst = ei + E;

    // workspace layout
    char* ws = (char*)d_ws;
    size_t off = 0;
    float* dinv = (float*)(ws + off); off = align_up(off + (size_t)NN * 4, 256);
    float* norm = (float*)(ws + off); off = align_up(off + (size_t)E * 4, 256);
    float* h    = (float*)(ws + off); off = align_up(off + (size_t)NN * HID * 4, 256);
    float* agg1 = (float*)(ws + off); off = align_up(off + (size_t)NN * HID * 4, 256);
    float* h2   = (float*)(ws + off); off = align_up(off + (size_t)NN * OUT_C * 4, 256);
    (void)ws_size;

    const int T = 256;

    // ---- normalization: deg -> dinv -> per-edge norm
    fill_val_kernel<<<(NN + T - 1) / T, T, 0, stream>>>(dinv, 1.0f, NN);
    deg_count_kernel<<<(E + T - 1) / T, T, 0, stream>>>(dst, dinv, E);
    rsqrt_ip_kernel<<<(NN + T - 1) / T, T, 0, stream>>>(dinv, NN);
    edge_norm_kernel<<<(E + T - 1) / T, T, 0, stream>>>(src, dst, dinv, norm, E);

    // ---- layer 1: h = x @ W1   (WMMA f32, 16x64 tiles per wave)
    {
        const int tiles = ((NN + 15) / 16) * (HID / 64);
        gemm_wmma_f32_kernel<false><<<(tiles + 3) / 4, 128, 0, stream>>>(
            x, W1, nullptr, h, NN, IN_C, HID);
    }
    // agg1 = self-loop term, then one-wave-per-edge scatter (f32 atomics)
    self_loop_init_kernel<<<((NN * (HID / 4)) + T - 1) / T, T, 0, stream>>>(
        (const float4*)h, dinv, nullptr, (float4*)agg1, NN, HID / 4);
    scatter_edges_kernel<<<(E + 7) / 8, T, 0, stream>>>(
        src, dst, norm, h, agg1, E, HID);

    // ---- layer 2: h2 = relu(agg1 + b1) @ W2   (bias+ReLU fused into A-load)
    {
        const int tiles = ((NN + 15) / 16) * (OUT_C / 64);
        gemm_wmma_f32_kernel<true><<<(tiles + 3) / 4, 128, 0, stream>>>(
            agg1, W2, b1, h2, NN, HID, OUT_C);
    }
    // out = self-loop term + b2, then scatter edges into d_out
    self_loop_init_kernel<<<((NN * (OUT_C / 4)) + T - 1) / T, T, 0, stream>>>(
        (const float4*)h2, dinv, (const float4*)b2, (float4*)out, NN, OUT_C / 4);
    scatter_edges_kernel<<<(E + 7) / 8, T, 0, stream>>>(
        src, dst, norm, h2, out, E, OUT_C);
}